// ReLU_62758062129325
// MI455X (gfx1250) — compile-verified
//
#include <hip/hip_runtime.h>

// Problem shape (fixed by the reference): B=16, N=4096, E=1025
#define NB    16
#define NN    4096
#define NE    1025
#define NW    1024              // weight columns (E-1)
#define ROWS  (NB * NN)         // 65536
#define TPB   256               // 8 wave32 waves
#define CHUNK 8                 // rows per block
#define STAGES 3                // triple-buffered async pipeline

// Full 32-lane butterfly sum via DS_SWIZZLE (wave32, group-of-32 xor mode).
// offset = xor<<10 | and_mask(0x1F)
__device__ __forceinline__ float wave_sum(float v) {
    int x;
    x = __builtin_amdgcn_ds_swizzle(__float_as_int(v), 0x401F); v += __int_as_float(x); // xor 16
    x = __builtin_amdgcn_ds_swizzle(__float_as_int(v), 0x201F); v += __int_as_float(x); // xor 8
    x = __builtin_amdgcn_ds_swizzle(__float_as_int(v), 0x101F); v += __int_as_float(x); // xor 4
    x = __builtin_amdgcn_ds_swizzle(__float_as_int(v), 0x081F); v += __int_as_float(x); // xor 2
    x = __builtin_amdgcn_ds_swizzle(__float_as_int(v), 0x041F); v += __int_as_float(x); // xor 1
    return v;
}

// Async global -> LDS copy of one 1024-float row: each lane moves 4 dwords at
// a 1024-float (4096-byte) stride, so every wave-instruction is UNIT-STRIDE across lanes
// (contiguous 128B per wave -> fully coalesced requests). The 24-bit
// INST_OFFSET is added to BOTH the global and LDS address per the CDNA5 async
// pseudocode, so one (lds,global) address pair serves all 4 chunks.
// th:TH_LOAD_NT — rows are streamed once (268MB arrays > 192MB L2).
__device__ __forceinline__ void async_copy_row(const float* __restrict__ g,
                                               const float* lds, int tid) {
    const float* gp = g + tid;
    unsigned lp = (unsigned)(unsigned long long)(lds + tid); // low 32 bits of LDS-aperture addr = LDS offset
    asm volatile("global_load_async_to_lds_b32 %0, %1, off th:TH_LOAD_NT"              :: "v"(lp), "v"(gp) : "memory");
    asm volatile("global_load_async_to_lds_b32 %0, %1, off offset:4096 th:TH_LOAD_NT"  :: "v"(lp), "v"(gp) : "memory");
    asm volatile("global_load_async_to_lds_b32 %0, %1, off offset:8192 th:TH_LOAD_NT"  :: "v"(lp), "v"(gp) : "memory");
    asm volatile("global_load_async_to_lds_b32 %0, %1, off offset:12288 th:TH_LOAD_NT" :: "v"(lp), "v"(gp) : "memory");
}

__global__ __launch_bounds__(TPB) void relu_dp_kernel(
    const float* __restrict__ l, const float* __restrict__ u,
    const float* __restrict__ in_lb, const float* __restrict__ in_ub,
    float* __restrict__ outL, float* __restrict__ outU,
    float* __restrict__ postL, float* __restrict__ postU)
{
    __shared__ __align__(16) float sL[STAGES][NW];
    __shared__ __align__(16) float sU[STAGES][NW];
    __shared__ float sred[TPB / 32][4];
    __shared__ float sbc[4]; // fac_l, fac_u, du

    const int tid  = threadIdx.x;
    const int lane = tid & 31;
    const int wid  = tid >> 5;

    // Interleaved column ownership: coalesced global access per instruction.
    const int c0 = tid;
    const int c1 = tid + 256;
    const int c2 = tid + 512;
    const int c3 = tid + 768;

    const size_t row0 = (size_t)blockIdx.x * CHUNK;

    // Interval midpoint/radius for this thread's 4 columns (shared by all rows).
    const float lb0 = in_lb[c0], lb1 = in_lb[c1], lb2 = in_lb[c2], lb3 = in_lb[c3];
    const float ub0 = in_ub[c0], ub1 = in_ub[c1], ub2 = in_ub[c2], ub3 = in_ub[c3];
    const float m0 = 0.5f * (lb0 + ub0), m1 = 0.5f * (lb1 + ub1);
    const float m2 = 0.5f * (lb2 + ub2), m3 = 0.5f * (lb3 + ub3);
    const float r0 = 0.5f * (ub0 - lb0), r1 = 0.5f * (ub1 - lb1);
    const float r2 = 0.5f * (ub2 - lb2), r3 = 0.5f * (ub3 - lb3);

    // Kick off async fill of stage 0 (8 outstanding b32 asyncs per wave).
    async_copy_row(l + row0 * NE, &sL[0][0], tid);
    async_copy_row(u + row0 * NE, &sU[0][0], tid);

    for (int i = 0; i < CHUNK; ++i) {
        const size_t row = row0 + i;
        const int st = i % STAGES;

        // Prefetch next row into stage (i+1)%3, then release the current row.
        // (Stage i+1 last held row i-2; two full barrier rounds separate its
        //  readers from this write, so the triple buffer is hazard-free.)
        if (i + 1 < CHUNK) {
            async_copy_row(l + (row + 1) * NE, &sL[(i + 1) % STAGES][0], tid);
            async_copy_row(u + (row + 1) * NE, &sU[(i + 1) % STAGES][0], tid);
            asm volatile("s_wait_asynccnt 8" ::: "memory");  // current row's 8 loads done (in-order)
        } else {
            asm volatile("s_wait_asynccnt 0" ::: "memory");
        }

        // Biases via normal loads (latency covered by the reduction below).
        float bl = 0.0f, bu = 0.0f;
        if (tid == 0) { bl = l[row * NE + NW]; bu = u[row * NE + NW]; }

        // Row data: LDS -> registers (kept live through the barriers for the rescale).
        const float l0 = sL[st][c0], l1 = sL[st][c1], l2 = sL[st][c2], l3 = sL[st][c3];
        const float u0 = sU[st][c0], u1 = sU[st][c1], u2 = sU[st][c2], u3 = sU[st][c3];

        // s = W.mid, t = |W|.rad  (|.| folds into the FMA as a source modifier)
        float sl = fmaf(l0, m0, fmaf(l1, m1, fmaf(l2, m2, l3 * m3)));
        float tl = fmaf(fabsf(l0), r0, fmaf(fabsf(l1), r1, fmaf(fabsf(l2), r2, fabsf(l3) * r3)));
        float su = fmaf(u0, m0, fmaf(u1, m1, fmaf(u2, m2, u3 * m3)));
        float tu = fmaf(fabsf(u0), r0, fmaf(fabsf(u1), r1, fmaf(fabsf(u2), r2, fabsf(u3) * r3)));

        sl = wave_sum(sl); tl = wave_sum(tl); su = wave_sum(su); tu = wave_sum(tu);
        if (lane == 0) { sred[wid][0] = sl; sred[wid][1] = tl; sred[wid][2] = su; sred[wid][3] = tu; }
        __syncthreads();

        if (tid == 0) {
            float Sl = 0.f, Tl = 0.f, Su = 0.f, Tu = 0.f;
            #pragma unroll
            for (int w = 0; w < TPB / 32; ++w) {
                Sl += sred[w][0]; Tl += sred[w][1]; Su += sred[w][2]; Tu += sred[w][3];
            }
            const float conc_lb = Sl - Tl + bl;   // pos(Wl).lb + neg(Wl).ub + bl
            const float max_lb  = Sl + Tl + bl;   // pos(Wl).ub + neg(Wl).lb + bl
            const float min_ub  = Su - Tu + bu;
            const float conc_ub = Su + Tu + bu;

            const bool inactive        = (conc_ub <= 0.0f);
            const bool unstable        = (conc_lb < 0.0f) && (conc_ub > 0.0f);
            const bool mostly_active   = unstable && (fabsf(conc_lb) <= fabsf(conc_ub));
            const bool mostly_inactive = unstable && !mostly_active;

            float fac_l;
            if (inactive || mostly_inactive) fac_l = 0.0f;
            else if (mostly_active)
                fac_l = (max_lb < 0.0f) ? 0.0f : (max_lb / (max_lb - conc_lb));
            else fac_l = 1.0f;

            const bool  zero_cross = unstable && (min_ub <= 0.0f);
            const float a_u   = zero_cross ? (conc_ub / (conc_ub - min_ub)) : 0.0f;
            const float fac_u = inactive ? 0.0f : (zero_cross ? a_u : 1.0f);
            const float du    = zero_cross ? (-a_u * min_ub) : 0.0f;  // zero_cross => !inactive

            sbc[0] = fac_l; sbc[1] = fac_u; sbc[2] = du;

            // Analytic post-concretization (l_new/u_new are scalar rescales of l/u):
            postL[row] = fac_l * conc_lb;
            postU[row] = inactive ? 0.0f : conc_ub;

            // Bias column of the outputs:
            __builtin_nontemporal_store(fac_l * bl, &outL[row * NE + NW]);
            __builtin_nontemporal_store(fmaf(fac_u, bu, du), &outU[row * NE + NW]);
        }
        __syncthreads();

        const float fl = sbc[0], fu = sbc[1];
        float* ol = outL + row * NE;
        float* ou = outU + row * NE;
        // Unit-stride per store instruction -> fully coalesced NT streams.
        __builtin_nontemporal_store(fl * l0, ol + c0);
        __builtin_nontemporal_store(fl * l1, ol + c1);
        __builtin_nontemporal_store(fl * l2, ol + c2);
        __builtin_nontemporal_store(fl * l3, ol + c3);
        __builtin_nontemporal_store(fu * u0, ou + c0);
        __builtin_nontemporal_store(fu * u1, ou + c1);
        __builtin_nontemporal_store(fu * u2, ou + c2);
        __builtin_nontemporal_store(fu * u3, ou + c3);
    }
}

extern "C" void kernel_launch(void* const* d_in, const int* in_sizes, int n_in,
                              void* d_out, int out_size, void* d_ws, size_t ws_size,
                              hipStream_t stream) {
    const float* l     = (const float*)d_in[0];
    const float* u     = (const float*)d_in[1];
    const float* in_lb = (const float*)d_in[2];
    const float* in_ub = (const float*)d_in[3];

    float* outL  = (float*)d_out;
    float* outU  = outL + (size_t)ROWS * NE;
    float* postL = outU + (size_t)ROWS * NE;
    float* postU = postL + ROWS;

    dim3 grid(ROWS / CHUNK);
    dim3 block(TPB);
    hipLaunchKernelGGL(relu_dp_kernel, grid, block, 0, stream,
                       l, u, in_lb, in_ub, outL, outU, postL, postU);
}